// EnergyDiffusionImputer_2731599200258
// MI455X (gfx1250) — compile-verified
//
#include <hip/hip_runtime.h>
#include <hip/hip_bf16.h>
#include <stdint.h>

// ---------------------------------------------------------------------------
// EnergyDiffusionImputer on gfx1250 (MI455X), fp32 WMMA (V_WMMA_F32_16X16X4_F32)
// + GLOBAL_LOAD_ASYNC_TO_LDS_B128 staging, dual-accumulator WMMA chains.
// ---------------------------------------------------------------------------
namespace {
constexpr int BATCH = 131072;
constexpr int DXc   = 256;
constexpr int DYc   = 32;
constexpr int Hc    = 128;
constexpr int STEPS_MAX = 20;
constexpr float LRc  = 0.1f;
constexpr float REGc = 0.01f;

constexpr int LDB  = 132;          // padded LDS row stride (floats) -> conflict free, 528B (16B mult)
constexpr int BUF  = 16 * LDB;     // one 16xH staging buffer (2112 floats)
constexpr int MISC = 160;          // lg(64) + dl(64) + tci(16) + pad
constexpr int WPW_GRAD = 4 * BUF + MISC;                 // per-wave floats, grad kernel
constexpr int W_LDS    = 2 * (Hc * LDB) + 2 * (Hc * 4);  // e_w2 + tr1wY + e_w3 + tr2w
constexpr int GRAD_LDS_BYTES = (W_LDS + 4 * WPW_GRAD) * 4;   // ~277 KB  (< 320 KB WGP LDS)

constexpr int XS      = 16 * 260;  // padded 16x256 x-tile (1040B rows, 16B mult)
constexpr int WPW_PRE = XS + 3 * BUF;
constexpr int PRE_LDS_BYTES = 4 * WPW_PRE * 4;               // ~168 KB
}

typedef float v2f __attribute__((ext_vector_type(2)));
typedef float v8f __attribute__((ext_vector_type(8)));

// D(16x16,f32) = A(16x4,f32) * B(4x16,f32) + C   -- CDNA5 fp32 WMMA
__device__ __forceinline__ v8f wmma_k4(v2f a, v2f b, v8f c) {
  return __builtin_amdgcn_wmma_f32_16x16x4_f32(false, a, false, b, (short)0, c,
                                               false, false);
}

__device__ __forceinline__ float sigf(float z) { return 1.f / (1.f + __expf(-z)); }

// CDNA5 async copy: 16B global -> LDS per lane, tracked by ASYNCcnt.
__device__ __forceinline__ void async_ld_b128(const float* lds_ptr, const float* gptr) {
  uint32_t lds_off = (uint32_t)(uintptr_t)lds_ptr;  // low 32b of LDS-aperture addr = LDS offset
  asm volatile("global_load_async_to_lds_b128 %0, %1, off"
               :: "v"(lds_off), "v"(gptr)
               : "memory");
}
__device__ __forceinline__ void wait_async0() {
  asm volatile("s_wait_asynccnt 0" ::: "memory");
}

// ---------------------------------------------------------------------------
// init: zero y (output accumulator) and the per-step norm slots
// ---------------------------------------------------------------------------
__global__ void edi_init(float* __restrict__ y, float* __restrict__ norms) {
  size_t i = (size_t)blockIdx.x * blockDim.x + threadIdx.x;
  if (i < (size_t)BATCH * DYc) y[i] = 0.f;
  if (i < 32) norms[i] = 0.f;
}

// ---------------------------------------------------------------------------
// precompute: hx_e = x@e_w1[:256]+e_b1 ;  base = hx_s@tr1w0 + temb@tr1w2 + th@tr1w3 + tr1b
// one wave handles 16 rows; 4 waves / block
// ---------------------------------------------------------------------------
__global__ __launch_bounds__(128) void edi_pre(
    const float* __restrict__ x, const int* __restrict__ t,
    const float* __restrict__ e_w1, const float* __restrict__ e_b1,
    const float* __restrict__ s_xw, const float* __restrict__ s_xb,
    const float* __restrict__ s_temb,
    const float* __restrict__ s_t1w, const float* __restrict__ s_t1b,
    const float* __restrict__ s_t2w, const float* __restrict__ s_t2b,
    const float* __restrict__ tr1w, const float* __restrict__ tr1b,
    float* __restrict__ hx_e, float* __restrict__ baseb) {
  extern __shared__ float sm[];
  const int lane = threadIdx.x & 31;
  const int wv   = threadIdx.x >> 5;
  const int r0   = (blockIdx.x * 4 + wv) * 16;
  float* xs = sm + wv * WPW_PRE;   // 16x256 (stride 260), later reused for th (stride 132)
  float* bs = xs + XS;             // hx_s staging
  float* bt = bs + BUF;            // temb staging
  float* bh = bt + BUF;            // silu(tau*t1w+t1b) staging
  const int m16  = lane & 15;
  const int koff = 2 * (lane >> 4);
  const int mh   = 8 * (lane >> 4);

  // stage x tile via async copies (16 rows x 64 16B-chunks)
  for (int i = lane; i < 16 * 64; i += 32) {
    int r = i >> 6, cc = (i & 63) * 4;
    async_ld_b128(xs + r * 260 + cc, x + (size_t)(r0 + r) * DXc + cc);
  }
  wait_async0();

  // hx_s = x @ s_xw + s_xb  -> bs
  for (int nt = 0; nt < 8; ++nt) {
    int n = nt * 16 + m16;
    v8f c0, c1 = {};
    float bias = s_xb[n];
#pragma unroll
    for (int j = 0; j < 8; ++j) c0[j] = bias;
    for (int kc = 0; kc < 64; kc += 2) {
      int kb = kc * 4 + koff;
      v2f a0 = *(const v2f*)(xs + m16 * 260 + kb);
      v2f b0; b0.x = s_xw[kb * Hc + n]; b0.y = s_xw[(kb + 1) * Hc + n];
      c0 = wmma_k4(a0, b0, c0);
      v2f a1 = *(const v2f*)(xs + m16 * 260 + kb + 4);
      v2f b1; b1.x = s_xw[(kb + 4) * Hc + n]; b1.y = s_xw[(kb + 5) * Hc + n];
      c1 = wmma_k4(a1, b1, c1);
    }
#pragma unroll
    for (int j = 0; j < 8; ++j) bs[(j + mh) * LDB + n] = c0[j] + c1[j];
  }
  // hx_e = x @ e_w1[:256] + e_b1  -> global ws
  for (int nt = 0; nt < 8; ++nt) {
    int n = nt * 16 + m16;
    v8f c0, c1 = {};
    float bias = e_b1[n];
#pragma unroll
    for (int j = 0; j < 8; ++j) c0[j] = bias;
    for (int kc = 0; kc < 64; kc += 2) {
      int kb = kc * 4 + koff;
      v2f a0 = *(const v2f*)(xs + m16 * 260 + kb);
      v2f b0; b0.x = e_w1[kb * Hc + n]; b0.y = e_w1[(kb + 1) * Hc + n];
      c0 = wmma_k4(a0, b0, c0);
      v2f a1 = *(const v2f*)(xs + m16 * 260 + kb + 4);
      v2f b1; b1.x = e_w1[(kb + 4) * Hc + n]; b1.y = e_w1[(kb + 5) * Hc + n];
      c1 = wmma_k4(a1, b1, c1);
    }
#pragma unroll
    for (int j = 0; j < 8; ++j) hx_e[(size_t)(r0 + j + mh) * Hc + n] = c0[j] + c1[j];
  }
  // elementwise: th_in = silu(tau*s_t1w+s_t1b) -> bh ; temb gather -> bt
  for (int i = lane; i < 16 * 128; i += 32) {
    int m = i >> 7, jj = i & 127;
    int tt = t[r0 + m];
    int tcv = tt < 0 ? 0 : tt;
    float tau = fmaxf((float)tcv * 1e-3f, 1e-6f);
    float z = tau * s_t1w[jj] + s_t1b[jj];
    bh[m * LDB + jj] = z * sigf(z);
    bt[m * LDB + jj] = s_temb[tcv * Hc + jj];
  }
  // th = th_in @ s_t2w + s_t2b  -> reuse xs (stride 132)
  for (int nt = 0; nt < 8; ++nt) {
    int n = nt * 16 + m16;
    v8f c0, c1 = {};
    float bias = s_t2b[n];
#pragma unroll
    for (int j = 0; j < 8; ++j) c0[j] = bias;
    for (int kc = 0; kc < 32; kc += 2) {
      int kb = kc * 4 + koff;
      v2f a0 = *(const v2f*)(bh + m16 * LDB + kb);
      v2f b0; b0.x = s_t2w[kb * Hc + n]; b0.y = s_t2w[(kb + 1) * Hc + n];
      c0 = wmma_k4(a0, b0, c0);
      v2f a1 = *(const v2f*)(bh + m16 * LDB + kb + 4);
      v2f b1; b1.x = s_t2w[(kb + 4) * Hc + n]; b1.y = s_t2w[(kb + 5) * Hc + n];
      c1 = wmma_k4(a1, b1, c1);
    }
#pragma unroll
    for (int j = 0; j < 8; ++j) xs[(j + mh) * LDB + n] = c0[j] + c1[j];
  }
  // base = hx_s@tr1w[0:128] + temb@tr1w[256:384] + th@tr1w[384:512] + tr1b
  for (int nt = 0; nt < 8; ++nt) {
    int n = nt * 16 + m16;
    v8f c0, c1 = {};
    float bias = tr1b[n];
#pragma unroll
    for (int j = 0; j < 8; ++j) c0[j] = bias;
    for (int kc = 0; kc < 32; kc += 2) {
      int kb = kc * 4 + koff;
      v2f a0 = *(const v2f*)(bs + m16 * LDB + kb);
      v2f b0; b0.x = tr1w[kb * Hc + n]; b0.y = tr1w[(kb + 1) * Hc + n];
      c0 = wmma_k4(a0, b0, c0);
      v2f a1 = *(const v2f*)(bs + m16 * LDB + kb + 4);
      v2f b1; b1.x = tr1w[(kb + 4) * Hc + n]; b1.y = tr1w[(kb + 5) * Hc + n];
      c1 = wmma_k4(a1, b1, c1);
    }
    for (int kc = 0; kc < 32; kc += 2) {
      int kb = kc * 4 + koff;
      v2f a0 = *(const v2f*)(bt + m16 * LDB + kb);
      v2f b0; b0.x = tr1w[(256 + kb) * Hc + n]; b0.y = tr1w[(257 + kb) * Hc + n];
      c0 = wmma_k4(a0, b0, c0);
      v2f a1 = *(const v2f*)(bt + m16 * LDB + kb + 4);
      v2f b1; b1.x = tr1w[(260 + kb) * Hc + n]; b1.y = tr1w[(261 + kb) * Hc + n];
      c1 = wmma_k4(a1, b1, c1);
    }
    for (int kc = 0; kc < 32; kc += 2) {
      int kb = kc * 4 + koff;
      v2f a0 = *(const v2f*)(xs + m16 * LDB + kb);
      v2f b0; b0.x = tr1w[(384 + kb) * Hc + n]; b0.y = tr1w[(385 + kb) * Hc + n];
      c0 = wmma_k4(a0, b0, c0);
      v2f a1 = *(const v2f*)(xs + m16 * LDB + kb + 4);
      v2f b1; b1.x = tr1w[(388 + kb) * Hc + n]; b1.y = tr1w[(389 + kb) * Hc + n];
      c1 = wmma_k4(a1, b1, c1);
    }
#pragma unroll
    for (int j = 0; j < 8; ++j) baseb[(size_t)(r0 + j + mh) * Hc + n] = c0[j] + c1[j];
  }
}

// ---------------------------------------------------------------------------
// per-step gradient: full fwd+bwd for 16 rows per wave, writes g and ||g||^2
// ---------------------------------------------------------------------------
__global__ __launch_bounds__(128) void edi_grad(
    const float* __restrict__ y, const int* __restrict__ t,
    const float* __restrict__ e_w1, const float* __restrict__ e_b2,
    const float* __restrict__ e_w2, const float* __restrict__ e_w3,
    const float* __restrict__ s_yw, const float* __restrict__ s_yb,
    const float* __restrict__ tr1w, const float* __restrict__ tr2w,
    const float* __restrict__ tr2b,
    const float* __restrict__ hx_e, const float* __restrict__ baseb,
    float* __restrict__ g_out, float* __restrict__ norm_slot,
    const int* __restrict__ steps_p, int step) {
  if (step >= *steps_p) return;
  extern __shared__ float sm[];
  float* ew2s  = sm;                    // [128][132] forward+transposed reads
  float* t1ys  = ew2s + Hc * LDB;       // tr1w rows 128..255, [128][132]
  float* ew3s  = t1ys + Hc * LDB;       // [128][4]
  float* tr2ws = ew3s + Hc * 4;         // [128][4]

  __builtin_prefetch(e_w2, 0, 3);
  __builtin_prefetch(tr1w + 128 * Hc, 0, 3);
  // async-stage shared weights: 128 rows x 32 16B-chunks each
  for (int i = threadIdx.x; i < 128 * 32; i += 128) {
    int r = i >> 5, cc = (i & 31) * 4;
    async_ld_b128(ew2s + r * LDB + cc, e_w2 + r * Hc + cc);
    async_ld_b128(t1ys + r * LDB + cc, tr1w + (size_t)(128 + r) * Hc + cc);
  }
  {
    int i = threadIdx.x;  // 128 threads x 1 chunk covers 512 floats
    async_ld_b128(ew3s + i * 4, e_w3 + i * 4);
    async_ld_b128(tr2ws + i * 4, tr2w + i * 4);
  }
  wait_async0();
  __syncthreads();

  const int lane = threadIdx.x & 31;
  const int wv   = threadIdx.x >> 5;
  const int r0   = (blockIdx.x * 4 + wv) * 16;
  float* b0 = sm + W_LDS + wv * WPW_GRAD;  // h1
  float* b1 = b0 + BUF;                    // h2
  float* b2 = b1 + BUF;                    // hy -> dzt -> dz1
  float* b3 = b2 + BUF;                    // zt -> dhy
  float* lg = b3 + BUF;                    // logits 16x4
  float* dl = lg + 64;                     // dlogits 16x4
  int*   tci = (int*)(dl + 64);            // class per row
  const int m16  = lane & 15;
  const int koff = 2 * (lane >> 4);
  const int mh   = 8 * (lane >> 4);

  // ---- z1 = hx_e + y@e_w1[256:288] ; h1 = relu -> b0
  for (int nt = 0; nt < 8; ++nt) {
    int n = nt * 16 + m16;
    v8f c0, c1 = {};
#pragma unroll
    for (int j = 0; j < 8; ++j) c0[j] = hx_e[(size_t)(r0 + j + mh) * Hc + n];
    for (int kc = 0; kc < 8; kc += 2) {
      int kb = kc * 4 + koff;
      v2f a0 = *(const v2f*)(y + (size_t)(r0 + m16) * DYc + kb);
      v2f b0f; b0f.x = e_w1[(DXc + kb) * Hc + n]; b0f.y = e_w1[(DXc + kb + 1) * Hc + n];
      c0 = wmma_k4(a0, b0f, c0);
      v2f a1 = *(const v2f*)(y + (size_t)(r0 + m16) * DYc + kb + 4);
      v2f b1f; b1f.x = e_w1[(DXc + kb + 4) * Hc + n]; b1f.y = e_w1[(DXc + kb + 5) * Hc + n];
      c1 = wmma_k4(a1, b1f, c1);
    }
#pragma unroll
    for (int j = 0; j < 8; ++j) b0[(j + mh) * LDB + n] = fmaxf(c0[j] + c1[j], 0.f);
  }
  // ---- z2 = h1@e_w2 + e_b2 ; h2 = relu -> b1
  for (int nt = 0; nt < 8; ++nt) {
    int n = nt * 16 + m16;
    v8f c0, c1 = {};
    float bias = e_b2[n];
#pragma unroll
    for (int j = 0; j < 8; ++j) c0[j] = bias;
    for (int kc = 0; kc < 32; kc += 2) {
      int kb = kc * 4 + koff;
      v2f a0 = *(const v2f*)(b0 + m16 * LDB + kb);
      v2f b0f; b0f.x = ew2s[kb * LDB + n]; b0f.y = ew2s[(kb + 1) * LDB + n];
      c0 = wmma_k4(a0, b0f, c0);
      v2f a1 = *(const v2f*)(b0 + m16 * LDB + kb + 4);
      v2f b1f; b1f.x = ew2s[(kb + 4) * LDB + n]; b1f.y = ew2s[(kb + 5) * LDB + n];
      c1 = wmma_k4(a1, b1f, c1);
    }
#pragma unroll
    for (int j = 0; j < 8; ++j) b1[(j + mh) * LDB + n] = fmaxf(c0[j] + c1[j], 0.f);
  }
  // ---- hy = y@s_yw + s_yb -> b2
  for (int nt = 0; nt < 8; ++nt) {
    int n = nt * 16 + m16;
    v8f c0, c1 = {};
    float bias = s_yb[n];
#pragma unroll
    for (int j = 0; j < 8; ++j) c0[j] = bias;
    for (int kc = 0; kc < 8; kc += 2) {
      int kb = kc * 4 + koff;
      v2f a0 = *(const v2f*)(y + (size_t)(r0 + m16) * DYc + kb);
      v2f b0f; b0f.x = s_yw[kb * Hc + n]; b0f.y = s_yw[(kb + 1) * Hc + n];
      c0 = wmma_k4(a0, b0f, c0);
      v2f a1 = *(const v2f*)(y + (size_t)(r0 + m16) * DYc + kb + 4);
      v2f b1f; b1f.x = s_yw[(kb + 4) * Hc + n]; b1f.y = s_yw[(kb + 5) * Hc + n];
      c1 = wmma_k4(a1, b1f, c1);
    }
#pragma unroll
    for (int j = 0; j < 8; ++j) b2[(j + mh) * LDB + n] = c0[j] + c1[j];
  }
  // ---- zt = base + hy@tr1w[128:256] -> b3
  for (int nt = 0; nt < 8; ++nt) {
    int n = nt * 16 + m16;
    v8f c0, c1 = {};
#pragma unroll
    for (int j = 0; j < 8; ++j) c0[j] = baseb[(size_t)(r0 + j + mh) * Hc + n];
    for (int kc = 0; kc < 32; kc += 2) {
      int kb = kc * 4 + koff;
      v2f a0 = *(const v2f*)(b2 + m16 * LDB + kb);
      v2f b0f; b0f.x = t1ys[kb * LDB + n]; b0f.y = t1ys[(kb + 1) * LDB + n];
      c0 = wmma_k4(a0, b0f, c0);
      v2f a1 = *(const v2f*)(b2 + m16 * LDB + kb + 4);
      v2f b1f; b1f.x = t1ys[(kb + 4) * LDB + n]; b1f.y = t1ys[(kb + 5) * LDB + n];
      c1 = wmma_k4(a1, b1f, c1);
    }
#pragma unroll
    for (int j = 0; j < 8; ++j) b3[(j + mh) * LDB + n] = c0[j] + c1[j];
  }
  // ---- logits = silu(zt)@tr2w + tr2b (cols 0..3), softmax -> dlogits
  {
    int n = m16;
    v8f c0, c1 = {};
    float bias = (n < 4) ? tr2b[n] : 0.f;
#pragma unroll
    for (int j = 0; j < 8; ++j) c0[j] = bias;
    for (int kc = 0; kc < 32; kc += 2) {
      int kb = kc * 4 + koff;
      float z0 = b3[m16 * LDB + kb], z1 = b3[m16 * LDB + kb + 1];
      v2f a0; a0.x = z0 * sigf(z0); a0.y = z1 * sigf(z1);
      v2f b0f;
      b0f.x = (n < 4) ? tr2ws[kb * 4 + n] : 0.f;
      b0f.y = (n < 4) ? tr2ws[(kb + 1) * 4 + n] : 0.f;
      c0 = wmma_k4(a0, b0f, c0);
      float z2 = b3[m16 * LDB + kb + 4], z3 = b3[m16 * LDB + kb + 5];
      v2f a1; a1.x = z2 * sigf(z2); a1.y = z3 * sigf(z3);
      v2f b1f;
      b1f.x = (n < 4) ? tr2ws[(kb + 4) * 4 + n] : 0.f;
      b1f.y = (n < 4) ? tr2ws[(kb + 5) * 4 + n] : 0.f;
      c1 = wmma_k4(a1, b1f, c1);
    }
    if (n < 4) {
#pragma unroll
      for (int j = 0; j < 8; ++j) lg[(j + mh) * 4 + n] = c0[j] + c1[j];
    }
  }
  if (lane < 16) {
    int m = lane;
    int tt = t[r0 + m];
    int tcv = tt < 0 ? 0 : tt;
    tci[m] = tcv;
    float vv = (tt >= 0) ? 1.f : 0.f;
    float l0 = lg[m * 4], l1 = lg[m * 4 + 1], l2 = lg[m * 4 + 2], l3 = lg[m * 4 + 3];
    float mx = fmaxf(fmaxf(l0, l1), fmaxf(l2, l3));
    float e0 = __expf(l0 - mx), e1 = __expf(l1 - mx), e2 = __expf(l2 - mx), e3 = __expf(l3 - mx);
    float inv = 1.f / (e0 + e1 + e2 + e3);
    dl[m * 4 + 0] = (e0 * inv - (tcv == 0 ? 1.f : 0.f)) * vv;
    dl[m * 4 + 1] = (e1 * inv - (tcv == 1 ? 1.f : 0.f)) * vv;
    dl[m * 4 + 2] = (e2 * inv - (tcv == 2 ? 1.f : 0.f)) * vv;
    dl[m * 4 + 3] = (e3 * inv - (tcv == 3 ? 1.f : 0.f)) * vv;
  }
  // ---- dzt = (dlogits@tr2w^T) * silu'(zt)  -> b2 (hy dead)
  for (int nt = 0; nt < 8; ++nt) {
    int n = nt * 16 + m16;
    v8f c = {};
    v2f a; a.x = dl[m16 * 4 + koff]; a.y = dl[m16 * 4 + koff + 1];
    v2f b = *(const v2f*)(tr2ws + n * 4 + koff);
    c = wmma_k4(a, b, c);
#pragma unroll
    for (int j = 0; j < 8; ++j) {
      float z = b3[(j + mh) * LDB + n];
      float s = sigf(z);
      b2[(j + mh) * LDB + n] = c[j] * (s + z * s * (1.f - s));
    }
  }
  // ---- dhy = dzt @ tr1wY^T -> b3 (zt dead)
  for (int nt = 0; nt < 8; ++nt) {
    int n = nt * 16 + m16;
    v8f c0 = {}, c1 = {};
    for (int kc = 0; kc < 32; kc += 2) {
      int kb = kc * 4 + koff;
      v2f a0 = *(const v2f*)(b2 + m16 * LDB + kb);
      v2f b0f = *(const v2f*)(t1ys + n * LDB + kb);
      c0 = wmma_k4(a0, b0f, c0);
      v2f a1 = *(const v2f*)(b2 + m16 * LDB + kb + 4);
      v2f b1f = *(const v2f*)(t1ys + n * LDB + kb + 4);
      c1 = wmma_k4(a1, b1f, c1);
    }
#pragma unroll
    for (int j = 0; j < 8; ++j) b3[(j + mh) * LDB + n] = c0[j] + c1[j];
  }
  // ---- dy_ce = dhy @ s_yw^T (2 tiles, kept live)
  v8f cy[2];
#pragma unroll
  for (int nt2 = 0; nt2 < 2; ++nt2) {
    int n = nt2 * 16 + m16;
    v8f c0 = {}, c1 = {};
    for (int kc = 0; kc < 32; kc += 2) {
      int kb = kc * 4 + koff;
      v2f a0 = *(const v2f*)(b3 + m16 * LDB + kb);
      v2f b0f = *(const v2f*)(s_yw + n * Hc + kb);
      c0 = wmma_k4(a0, b0f, c0);
      v2f a1 = *(const v2f*)(b3 + m16 * LDB + kb + 4);
      v2f b1f = *(const v2f*)(s_yw + n * Hc + kb + 4);
      c1 = wmma_k4(a1, b1f, c1);
    }
#pragma unroll
    for (int j = 0; j < 8; ++j) c0[j] += c1[j];
    cy[nt2] = c0;
  }
  // ---- dz1 = (dz2 @ e_w2^T) * (h1>0), dz2 built on the fly -> b2
  for (int nt = 0; nt < 8; ++nt) {
    int n = nt * 16 + m16;
    v8f c0 = {}, c1 = {};
    int cls = tci[m16];
    for (int kc = 0; kc < 32; kc += 2) {
      int kb = kc * 4 + koff;
      v2f a0;
      a0.x = (b1[m16 * LDB + kb]     > 0.f) ? ew3s[kb * 4 + cls]       : 0.f;
      a0.y = (b1[m16 * LDB + kb + 1] > 0.f) ? ew3s[(kb + 1) * 4 + cls] : 0.f;
      v2f b0f = *(const v2f*)(ew2s + n * LDB + kb);
      c0 = wmma_k4(a0, b0f, c0);
      v2f a1;
      a1.x = (b1[m16 * LDB + kb + 4] > 0.f) ? ew3s[(kb + 4) * 4 + cls] : 0.f;
      a1.y = (b1[m16 * LDB + kb + 5] > 0.f) ? ew3s[(kb + 5) * 4 + cls] : 0.f;
      v2f b1f = *(const v2f*)(ew2s + n * LDB + kb + 4);
      c1 = wmma_k4(a1, b1f, c1);
    }
#pragma unroll
    for (int j = 0; j < 8; ++j) {
      float h1v = b0[(j + mh) * LDB + n];
      b2[(j + mh) * LDB + n] = (h1v > 0.f) ? (c0[j] + c1[j]) : 0.f;
    }
  }
  // ---- dy_e = dz1 @ e_w1y^T, accumulate into cy; then g = cy + 2*REG*y
#pragma unroll
  for (int nt2 = 0; nt2 < 2; ++nt2) {
    int n = nt2 * 16 + m16;
    v8f c0 = cy[nt2], c1 = {};
    for (int kc = 0; kc < 32; kc += 2) {
      int kb = kc * 4 + koff;
      v2f a0 = *(const v2f*)(b2 + m16 * LDB + kb);
      v2f b0f = *(const v2f*)(e_w1 + (size_t)(DXc + n) * Hc + kb);
      c0 = wmma_k4(a0, b0f, c0);
      v2f a1 = *(const v2f*)(b2 + m16 * LDB + kb + 4);
      v2f b1f = *(const v2f*)(e_w1 + (size_t)(DXc + n) * Hc + kb + 4);
      c1 = wmma_k4(a1, b1f, c1);
    }
#pragma unroll
    for (int j = 0; j < 8; ++j) c0[j] += c1[j];
    cy[nt2] = c0;
  }
  float lsum = 0.f;
#pragma unroll
  for (int nt2 = 0; nt2 < 2; ++nt2) {
    int n = nt2 * 16 + m16;
#pragma unroll
    for (int j = 0; j < 8; ++j) {
      size_t row = (size_t)(r0 + j + mh);
      float yv = y[row * DYc + n];
      float gv = cy[nt2][j] + 2.f * REGc * yv;
      g_out[row * DYc + n] = gv;
      lsum += gv * gv;
    }
  }
  for (int off = 16; off > 0; off >>= 1) lsum += __shfl_xor(lsum, off, 32);
  if (lane == 0) atomicAdd(norm_slot, lsum);
}

// ---------------------------------------------------------------------------
// update: y -= LR*g unless done (||g|| < 1e-3) or step >= steps
// ---------------------------------------------------------------------------
__global__ void edi_update(float* __restrict__ y, const float* __restrict__ g,
                           const float* __restrict__ norms,
                           const int* __restrict__ steps_p, int step) {
  if (step >= *steps_p) return;
  if (sqrtf(norms[step]) < 1e-3f) return;
  size_t i = (size_t)blockIdx.x * blockDim.x + threadIdx.x;
  if (i < (size_t)BATCH * DYc) y[i] -= LRc * g[i];
}

// ---------------------------------------------------------------------------
extern "C" void kernel_launch(void* const* d_in, const int* in_sizes, int n_in,
                              void* d_out, int out_size, void* d_ws, size_t ws_size,
                              hipStream_t stream) {
  (void)in_sizes; (void)n_in; (void)out_size; (void)ws_size;
  const float* x      = (const float*)d_in[0];
  const int*   t      = (const int*)d_in[1];
  const float* e_w1   = (const float*)d_in[2];
  const float* e_b1   = (const float*)d_in[3];
  const float* e_w2   = (const float*)d_in[4];
  const float* e_b2   = (const float*)d_in[5];
  const float* e_w3   = (const float*)d_in[6];
  const float* s_xw   = (const float*)d_in[8];
  const float* s_xb   = (const float*)d_in[9];
  const float* s_yw   = (const float*)d_in[10];
  const float* s_yb   = (const float*)d_in[11];
  const float* s_temb = (const float*)d_in[12];
  const float* s_t1w  = (const float*)d_in[13];
  const float* s_t1b  = (const float*)d_in[14];
  const float* s_t2w  = (const float*)d_in[15];
  const float* s_t2b  = (const float*)d_in[16];
  const float* tr1w   = (const float*)d_in[17];
  const float* tr1b   = (const float*)d_in[18];
  const float* tr2w   = (const float*)d_in[19];
  const float* tr2b   = (const float*)d_in[20];
  const int*   steps  = (const int*)d_in[21];

  float* yout  = (float*)d_out;
  float* ws    = (float*)d_ws;
  float* hx_e  = ws;                              // B x 128
  float* baseb = hx_e + (size_t)BATCH * Hc;       // B x 128
  float* g     = baseb + (size_t)BATCH * Hc;      // B x 32
  float* norms = g + (size_t)BATCH * DYc;         // 32 slots

  const int rowBlocks = BATCH / 64;               // 16 rows/wave * 4 waves
  edi_init<<<(BATCH * DYc + 255) / 256, 256, 0, stream>>>(yout, norms);
  edi_pre<<<rowBlocks, 128, PRE_LDS_BYTES, stream>>>(
      x, t, e_w1, e_b1, s_xw, s_xb, s_temb, s_t1w, s_t1b, s_t2w, s_t2b,
      tr1w, tr1b, hx_e, baseb);
  for (int s = 0; s < STEPS_MAX; ++s) {
    edi_grad<<<rowBlocks, 128, GRAD_LDS_BYTES, stream>>>(
        yout, t, e_w1, e_b2, e_w2, e_w3, s_yw, s_yb, tr1w, tr2w, tr2b,
        hx_e, baseb, g, norms + s, steps, s);
    edi_update<<<(BATCH * DYc + 255) / 256, 256, 0, stream>>>(yout, g, norms, steps, s);
  }
}